// GCN_layer1_31739808318041
// MI455X (gfx1250) — compile-verified
//
#include <hip/hip_runtime.h>
#include <math.h>

#define DIM     128
#define NNODES  100000
#define NEDGES  640000
#define NTILES  (NNODES / 16)   // 6250 exactly
#define SLOPE   0.01f

typedef __attribute__((ext_vector_type(2))) float v2f;
typedef __attribute__((ext_vector_type(8))) float v8f;

// workspace layout (float indices)
#define WS_UA   0              // u_a[128]
#define WS_UB   128            // u_b[128]
#define WS_CA   256            // c_a = b_lin . w_a
#define WS_CB   257            // c_b = b_lin . w_b + b_att
#define WS_MAX  258            // encoded-uint running max
#define WS_SUM  259            // softmax denominator
#define WS_ZERO 384            // 128 zero floats (B operand for unused WMMA columns)
#define WS_AS   512            // aS[NNODES]
#define WS_AD   (512 + NNODES) // aD[NNODES]

// ---- orderable float <-> uint encoding for atomic max ----
__device__ __forceinline__ unsigned fenc(float f) {
    unsigned u = __float_as_uint(f);
    return (u >> 31) ? ~u : (u | 0x80000000u);
}
__device__ __forceinline__ float fdec(unsigned e) {
    unsigned u = (e >> 31) ? (e & 0x7FFFFFFFu) : ~e;
    return __uint_as_float(u);
}

__device__ __forceinline__ float edge_logit(int e, const int* __restrict__ src,
                                            const int* __restrict__ dst,
                                            const float* ws) {
    float l = ws[WS_AS + src[e]] + ws[WS_AD + dst[e]];
    return (l >= 0.f) ? l : SLOPE * l;   // leaky relu (b_att folded into WS_CB)
}

// ---------------- kernel 1: tiny precompute ----------------
// u_a = W_lin @ W_att[0:128], u_b = W_lin @ W_att[128:256]; biases; zero pad; init reductions.
__global__ void k_prep(const float* __restrict__ W_lin, const float* __restrict__ b_lin,
                       const float* __restrict__ W_att, const float* __restrict__ b_att,
                       float* ws) {
    int k = threadIdx.x;   // 0..127
    float sa = 0.f, sb = 0.f;
    for (int j = 0; j < DIM; ++j) {
        float w = W_lin[k * DIM + j];
        sa = fmaf(w, W_att[j], sa);
        sb = fmaf(w, W_att[DIM + j], sb);
    }
    ws[WS_UA + k] = sa;
    ws[WS_UB + k] = sb;
    ws[WS_ZERO + k] = 0.f;                       // zero B column for lanes n >= 2
    if (k == 0) {
        float ca = 0.f, cb = 0.f;
        for (int j = 0; j < DIM; ++j) {
            ca = fmaf(b_lin[j], W_att[j], ca);
            cb = fmaf(b_lin[j], W_att[DIM + j], cb);
        }
        ws[WS_CA] = ca;
        ws[WS_CB] = cb + b_att[0];
        ((unsigned*)ws)[WS_MAX] = fenc(-INFINITY);   // 0x007FFFFF
        ws[WS_SUM] = 0.f;
    }
}

// ---------------- kernel 2: node scores via WMMA f32 16x16x4 ----------------
// aS/aD = (h @ [u_a u_b 0..0]) + bias; one wave per 16-node tile, K-loop of 32 WMMAs.
// B loads are branchless: lanes n>=2 read from a zeroed pad region, so the loop
// body is pure global_load_b64 + v_wmma with no exec-mask manipulation.
__global__ void __launch_bounds__(256) k_scores(const float* __restrict__ h, float* ws) {
    const int lane = threadIdx.x & 31;
    const int wave = threadIdx.x >> 5;
    const int tile = blockIdx.x * 8 + wave;
    if (tile >= NTILES) return;                 // wave-uniform -> EXEC all-ones below
    const int m    = lane & 15;                 // A: row / B,C,D: column
    const int half = lane >> 4;
    const float* hrow = h + (size_t)(tile * 16 + m) * DIM;
    const float* u = ws + ((m == 0) ? WS_UA : ((m == 1) ? WS_UB : WS_ZERO));

    v8f c = {0.f, 0.f, 0.f, 0.f, 0.f, 0.f, 0.f, 0.f};
    #pragma unroll
    for (int kk = 0; kk < 32; ++kk) {
        const int k = 4 * kk + 2 * half;
        v2f a;                                  // A 16x4: VGPR0->K=k, VGPR1->K=k+1
        a.x = hrow[k];
        a.y = hrow[k + 1];
        v2f b;                                  // B 4x16: col 0 = u_a, col 1 = u_b, rest 0
        b.x = u[k];
        b.y = u[k + 1];
        c = __builtin_amdgcn_wmma_f32_16x16x4_f32(false, a, false, b,
                                                  (short)0, c, false, false);
    }
    // D layout: lane n, VGPR i -> element (M = i + 8*half, N = n)
    if (m < 2) {
        const float bias = (m == 0) ? ws[WS_CA] : ws[WS_CB];
        float* outp = ws + ((m == 0) ? WS_AS : WS_AD) + tile * 16 + half * 8;
        #pragma unroll
        for (int i = 0; i < 8; ++i) outp[i] = c[i] + bias;
    }
}

// ---------------- kernel 3: global max of leaky-relu logits ----------------
__global__ void k_max(const int* __restrict__ src, const int* __restrict__ dst, float* ws) {
    __shared__ unsigned red[256];
    const int e = blockIdx.x * 256 + threadIdx.x;
    unsigned v = 0u;                             // minimum of the encoding
    if (e < NEDGES) v = fenc(edge_logit(e, src, dst, ws));
    red[threadIdx.x] = v;
    __syncthreads();
    for (int s = 128; s > 0; s >>= 1) {
        if (threadIdx.x < s) {
            unsigned o = red[threadIdx.x + s];
            if (o > red[threadIdx.x]) red[threadIdx.x] = o;
        }
        __syncthreads();
    }
    if (threadIdx.x == 0) atomicMax(((unsigned*)ws) + WS_MAX, red[0]);
}

// ---------------- kernel 4: softmax denominator ----------------
__global__ void k_sum(const int* __restrict__ src, const int* __restrict__ dst, float* ws) {
    __shared__ float red[256];
    const int e = blockIdx.x * 256 + threadIdx.x;
    const float maxv = fdec(((const unsigned*)ws)[WS_MAX]);
    float v = 0.f;
    if (e < NEDGES) v = __expf(edge_logit(e, src, dst, ws) - maxv);
    red[threadIdx.x] = v;
    __syncthreads();
    for (int s = 128; s > 0; s >>= 1) {
        if (threadIdx.x < s) red[threadIdx.x] += red[threadIdx.x + s];
        __syncthreads();
    }
    if (threadIdx.x == 0) atomicAdd(ws + WS_SUM, red[0]);
}

// ---------------- kernel 5: weighted scatter-add of messages ----------------
// one wave per edge; lane covers 4 of 128 dims via float4 + 4 f32 atomics.
__global__ void __launch_bounds__(256) k_scatter(const float* __restrict__ h,
                                                 const int* __restrict__ src,
                                                 const int* __restrict__ dst,
                                                 const float* ws, float* out) {
    const int lane = threadIdx.x & 31;
    const int wave = threadIdx.x >> 5;
    const int e = blockIdx.x * 8 + wave;
    if (e >= NEDGES) return;
    const int s = src[e], d = dst[e];
    const float maxv   = fdec(((const unsigned*)ws)[WS_MAX]);
    const float invsum = 1.f / ws[WS_SUM];
    float l = ws[WS_AS + s] + ws[WS_AD + d];
    l = (l >= 0.f) ? l : SLOPE * l;
    const float w = __expf(l - maxv) * invsum;

    const float4 hv = *(const float4*)(h + (size_t)s * DIM + lane * 4);
    float* o = out + (size_t)d * DIM + lane * 4;
    atomicAdd(o + 0, w * hv.x);
    atomicAdd(o + 1, w * hv.y);
    atomicAdd(o + 2, w * hv.z);
    atomicAdd(o + 3, w * hv.w);
}

// ---------------- kernel 6: in-place ReLU ----------------
__global__ void k_relu(float* out) {
    const size_t i = (size_t)blockIdx.x * 256 + threadIdx.x;
    if (i < (size_t)NNODES * DIM) out[i] = fmaxf(out[i], 0.f);
}

extern "C" void kernel_launch(void* const* d_in, const int* in_sizes, int n_in,
                              void* d_out, int out_size, void* d_ws, size_t ws_size,
                              hipStream_t stream) {
    const float* h     = (const float*)d_in[0];
    const int*   src   = (const int*)  d_in[1];
    const int*   dst   = (const int*)  d_in[2];
    const float* W_lin = (const float*)d_in[3];
    const float* b_lin = (const float*)d_in[4];
    const float* W_att = (const float*)d_in[5];
    const float* b_att = (const float*)d_in[6];
    float* out = (float*)d_out;
    float* ws  = (float*)d_ws;

    (void)hipMemsetAsync(d_out, 0, (size_t)NNODES * DIM * sizeof(float), stream);
    k_prep   <<<1, 128, 0, stream>>>(W_lin, b_lin, W_att, b_att, ws);
    k_scores <<<(NTILES + 7) / 8, 256, 0, stream>>>(h, ws);
    k_max    <<<NEDGES / 256, 256, 0, stream>>>(src, dst, ws);
    k_sum    <<<NEDGES / 256, 256, 0, stream>>>(src, dst, ws);
    k_scatter<<<NEDGES / 8, 256, 0, stream>>>(h, src, dst, ws, out);
    k_relu   <<<(NNODES * DIM) / 256, 256, 0, stream>>>(out);
}